// drebin_59708635349245
// MI455X (gfx1250) — compile-verified
//
#include <hip/hip_runtime.h>
#include <math.h>

#define NQ    16
#define DIM   65536
#define BATCH 64
#define NCLS  23

typedef __attribute__((ext_vector_type(2))) float v2f;
typedef __attribute__((ext_vector_type(8))) float v8f;

#if defined(__has_builtin)
#if __has_builtin(__builtin_amdgcn_wmma_f32_16x16x4_f32)
#define HAVE_WMMA_F32X4 1
#endif
#endif

__device__ __forceinline__ float2 cmul(float2 a, float2 b) {
  return make_float2(__builtin_fmaf(a.x, b.x, -a.y * b.y),
                     __builtin_fmaf(a.x, b.y,  a.y * b.x));
}
__device__ __forceinline__ float2 cadd(float2 a, float2 b) {
  return make_float2(a.x + b.x, a.y + b.y);
}

// ---------------------------------------------------------------------------
// Precompute fused 1q matrices  M = Rot(w[q]) * RX(x[b,q])  for both layers,
// plus CRX (cos, sin) half-angle params.
__global__ void k_mats(const float* __restrict__ x,
                       const float* __restrict__ w0, const float* __restrict__ x0,
                       const float* __restrict__ w1, const float* __restrict__ x1,
                       float2* __restrict__ mats, float* __restrict__ crxp) {
  unsigned idx = blockIdx.x * 256u + threadIdx.x;              // 2048 threads
  if (idx < 2048u) {
    unsigned layer = idx >> 10, rem = idx & 1023u;
    unsigned b = rem >> 4, q = rem & 15u;
    const float* w = layer ? w1 : w0;

    // RX(theta): [[c, -i s],[-i s, c]], half angle
    float th_rx = x[b * NQ + q] * 0.5f;
    float cr = cosf(th_rx), sr = sinf(th_rx);
    float2 r0 = make_float2(cr, 0.0f);      // diag
    float2 r1 = make_float2(0.0f, -sr);     // off-diag

    float phi = w[q * 3 + 0], th = w[q * 3 + 1], om = w[q * 3 + 2];
    float ct = cosf(th * 0.5f), stt = sinf(th * 0.5f);
    float ap = 0.5f * (phi + om), am = 0.5f * (phi - om);
    float cap = cosf(ap), sap = sinf(ap), cam = cosf(am), sam = sinf(am);
    float2 ra = make_float2( cap * ct, -sap * ct);   //  e^{-i(phi+om)/2} ct
    float2 rb = make_float2(-cam * stt, -sam * stt); //  -e^{ i(phi-om)/2} st
    float2 rc = make_float2( cam * stt, -sam * stt); //   e^{-i(phi-om)/2} st
    float2 rd = make_float2( cap * ct,  sap * ct);   //   e^{ i(phi+om)/2} ct

    // M = Rot @ RX
    float2 m00 = cadd(cmul(ra, r0), cmul(rb, r1));
    float2 m01 = cadd(cmul(ra, r1), cmul(rb, r0));
    float2 m10 = cadd(cmul(rc, r0), cmul(rd, r1));
    float2 m11 = cadd(cmul(rc, r1), cmul(rd, r0));
    float2* M = mats + ((size_t)idx << 2);
    M[0] = m00; M[1] = m01; M[2] = m10; M[3] = m11;
  }
  if (idx < 32u) {                                              // CRX params
    unsigned layer = idx >> 4, q = idx & 15u;
    float th = (layer ? x1 : x0)[q] * 0.5f;
    crxp[idx * 2 + 0] = cosf(th);
    crxp[idx * 2 + 1] = sinf(th);
  }
}

// ---------------------------------------------------------------------------
// Shared-tile gate helpers (tile = 2^11 complex amplitudes in LDS).
__device__ __forceinline__ void tile_1q(float2* tile, unsigned lbit,
                                        float2 m00, float2 m01, float2 m10, float2 m11,
                                        unsigned t) {
  unsigned stride = 1u << lbit, mask = stride - 1u;
#pragma unroll
  for (int r = 0; r < 4; ++r) {                               // 1024 pairs / 256 thr
    unsigned tp = t + 256u * r;
    unsigned i0 = ((tp & ~mask) << 1) | (tp & mask);
    float2 s0 = tile[i0], s1 = tile[i0 + stride];
    tile[i0]          = cadd(cmul(m00, s0), cmul(m01, s1));
    tile[i0 + stride] = cadd(cmul(m10, s0), cmul(m11, s1));
  }
}

__device__ __forceinline__ void tile_crx(float2* tile, unsigned cb, unsigned tb,
                                         float c, float s, unsigned t) {
  unsigned blo = cb < tb ? cb : tb;
  unsigned bhi = cb < tb ? tb : cb;
  unsigned lom = (1u << blo) - 1u, him = (1u << bhi) - 1u;
#pragma unroll
  for (int r = 0; r < 2; ++r) {                               // 512 pairs / 256 thr
    unsigned p = t + 256u * r;
    unsigned i = ((p & ~lom) << 1) | (p & lom);               // insert 0 at blo
    i = ((i & ~him) << 1) | (i & him);                        // insert 0 at bhi
    unsigned i0 = i | (1u << cb);                             // ctrl=1, tgt=0
    unsigned i1 = i0 | (1u << tb);                            // ctrl=1, tgt=1
    float2 t0 = tile[i0], t1 = tile[i1];
    // t0' = c*t0 - i s*t1 ; t1' = -i s*t0 + c*t1
    tile[i0] = make_float2(__builtin_fmaf(c, t0.x,  s * t1.y),
                           __builtin_fmaf(c, t0.y, -s * t1.x));
    tile[i1] = make_float2(__builtin_fmaf(c, t1.x,  s * t0.y),
                           __builtin_fmaf(c, t1.y, -s * t0.x));
  }
}

// ---------------------------------------------------------------------------
// Pass 1: contiguous tile over global bits 0..10.
// Applies fused 1q gates wires 6..15 (local bits 9..0).
// init!=0: materialize |0...0> directly in LDS (state delta lies in tile hi==0),
// skipping the global read entirely (also replaces a separate init kernel).
__global__ void k_low1q(float2* __restrict__ st, const float2* __restrict__ mats,
                        int layer, int init) {
  __shared__ float2 tile[2048];
  unsigned wg = blockIdx.x;                                    // 64 b * 32 hi-groups
  unsigned b = wg >> 5, hi = wg & 31u;
  size_t base = ((size_t)b << 16) | ((size_t)hi << 11);
  unsigned t = threadIdx.x;
  if (init) {
#pragma unroll
    for (int j = 0; j < 8; ++j) {
      unsigned l = t + 256u * j;
      tile[l] = make_float2((hi == 0u && l == 0u) ? 1.0f : 0.0f, 0.0f);
    }
  } else {
#pragma unroll
    for (int j = 0; j < 8; ++j) tile[t + 256u * j] = st[base + t + 256u * j];
  }
  __syncthreads();

  for (int q = 6; q < 16; ++q) {
    const float2* M = mats + ((((size_t)layer * BATCH + b) * NQ + q) << 2);
    tile_1q(tile, 15u - (unsigned)q, M[0], M[1], M[2], M[3], t);
    __syncthreads();
  }
#pragma unroll
  for (int j = 0; j < 8; ++j) st[base + t + 256u * j] = tile[t + 256u * j];
}

// ---------------------------------------------------------------------------
// Pass 2: gathered tile over global bits {0..3, 9..15}; bits 4..8 fixed.
// local l (11b): l[10:4] -> global bits 15..9 ; l[3:0] -> global bits 3..0.
// Applies 1q wires 0..5 (local bit 10-q) then CRX q=0..5 (local (10-q, 9-q)).
__global__ void k_high(float2* __restrict__ st, const float2* __restrict__ mats,
                       const float* __restrict__ crxp, int layer) {
  __shared__ float2 tile[2048];
  unsigned wg = blockIdx.x;                                    // 64 b * 32 mid-groups
  unsigned b = wg >> 5, mid = wg & 31u;
  size_t sbase = (size_t)b << 16;
  unsigned t = threadIdx.x;
#pragma unroll
  for (int j = 0; j < 8; ++j) {
    unsigned l = t + 256u * j;
    unsigned g = ((l >> 4) << 9) | (mid << 4) | (l & 15u);
    tile[l] = st[sbase + g];
  }
  __syncthreads();

  for (int q = 0; q < 6; ++q) {
    const float2* M = mats + ((((size_t)layer * BATCH + b) * NQ + q) << 2);
    tile_1q(tile, 10u - (unsigned)q, M[0], M[1], M[2], M[3], t);
    __syncthreads();
  }
  for (int q = 0; q < 6; ++q) {
    float c = crxp[(layer * NQ + q) * 2 + 0];
    float s = crxp[(layer * NQ + q) * 2 + 1];
    tile_crx(tile, 10u - (unsigned)q, 9u - (unsigned)q, c, s, t);
    __syncthreads();
  }

#pragma unroll
  for (int j = 0; j < 8; ++j) {
    unsigned l = t + 256u * j;
    unsigned g = ((l >> 4) << 9) | (mid << 4) | (l & 15u);
    st[sbase + g] = tile[l];
  }
}

// ---------------------------------------------------------------------------
// Pass 3: gathered tile over global bits {0..9, 15}; bits 10..14 fixed.
// local l (11b): l[10] -> global bit 15 ; l[9:0] -> global bits 9..0.
// Applies CRX q=6..14 (local bits (15-q, 14-q), all <= 9) then the
// ring-closing CRX q=15: ctrl wire15 = bit0 -> local 0, tgt wire0 = bit15 -> local 10.
__global__ void k_lowwrap(float2* __restrict__ st, const float* __restrict__ crxp,
                          int layer) {
  __shared__ float2 tile[2048];
  unsigned wg = blockIdx.x;                                    // 64 b * 32 mid-groups
  unsigned b = wg >> 5, mid = wg & 31u;
  size_t sbase = (size_t)b << 16;
  unsigned t = threadIdx.x;
#pragma unroll
  for (int j = 0; j < 8; ++j) {
    unsigned l = t + 256u * j;
    unsigned g = ((l >> 10) << 15) | (mid << 10) | (l & 1023u);
    tile[l] = st[sbase + g];
  }
  __syncthreads();

  for (int q = 6; q < 15; ++q) {
    float c = crxp[(layer * NQ + q) * 2 + 0];
    float s = crxp[(layer * NQ + q) * 2 + 1];
    tile_crx(tile, 15u - (unsigned)q, 14u - (unsigned)q, c, s, t);
    __syncthreads();
  }
  {
    float c = crxp[(layer * NQ + 15) * 2 + 0];
    float s = crxp[(layer * NQ + 15) * 2 + 1];
    tile_crx(tile, 0u, 10u, c, s, t);                          // ctrl local 0, tgt local 10
    __syncthreads();
  }

#pragma unroll
  for (int j = 0; j < 8; ++j) {
    unsigned l = t + 256u * j;
    unsigned g = ((l >> 10) << 15) | (mid << 10) | (l & 1023u);
    st[sbase + g] = tile[l];
  }
}

// ---------------------------------------------------------------------------
// feats(b, w) = sum_k |state[b,k]|^2 * sign(k, w),  sign = 1-2*((k>>(15-w))&1)
// GEMM M=16(batch tile) x N=16(wires) x K=65536 via V_WMMA_F32_16X16X4_F32.
__global__ void k_feats(const float2* __restrict__ st, float* __restrict__ feats) {
  int m0 = blockIdx.x * 16;                                    // batch tile
#if HAVE_WMMA_F32X4
  __shared__ float red[8 * 256];                               // 8 partial 16x16 tiles
  int lane = threadIdx.x & 31;
  int wv   = threadIdx.x >> 5;
  int row  = m0 + (lane & 15);                                 // A-fragment M
  int colw = lane & 15;                                        // B-fragment N (wire)
  int khalf = lane >> 4;                                       // K half-select
  int shiftW = 15 - colw;
  const float2* srow = st + ((size_t)row << 16);

  v8f acc = {0.f, 0.f, 0.f, 0.f, 0.f, 0.f, 0.f, 0.f};
  int k0 = wv * (DIM / 8);
  for (int k = k0; k < k0 + DIM / 8; k += 4) {
    int ke = k + (khalf << 1);                                 // this lane's K pair
    float2 a0 = srow[ke], a1 = srow[ke + 1];
    v2f a; a[0] = __builtin_fmaf(a0.x, a0.x, a0.y * a0.y);
           a[1] = __builtin_fmaf(a1.x, a1.x, a1.y * a1.y);
    v2f bm; bm[0] = ((ke >> shiftW) & 1) ? -1.0f : 1.0f;
            bm[1] = (((ke + 1) >> shiftW) & 1) ? -1.0f : 1.0f;
    acc = __builtin_amdgcn_wmma_f32_16x16x4_f32(
        false, a, false, bm, (short)0, acc, false, false);
  }
#pragma unroll
  for (int r = 0; r < 8; ++r) red[wv * 256 + lane * 8 + r] = acc[r];
  __syncthreads();

  // tid -> (L = tid/8, r = tid%8): C layout: M = r + 8*(L>>4), N = L&15
  int t = threadIdx.x;
  int L = t >> 3, r = t & 7;
  float s = 0.0f;
#pragma unroll
  for (int wvi = 0; wvi < 8; ++wvi) s += red[wvi * 256 + L * 8 + r];
  int M = r + ((L >> 4) << 3);
  int N = L & 15;
  feats[(m0 + M) * NQ + N] = s;
#else
  // Scalar fallback (keeps correctness if the x4_f32 builtin is absent).
  int t = threadIdx.x;                                         // 256 threads
  int row = m0 + (t >> 4);
  int wire = t & 15;
  int sh = 15 - wire;
  const float2* srow = st + ((size_t)row << 16);
  float s = 0.0f;
  for (int k = 0; k < DIM; ++k) {
    float2 a = srow[k];
    float pr = __builtin_fmaf(a.x, a.x, a.y * a.y);
    s += ((k >> sh) & 1) ? -pr : pr;
  }
  feats[row * NQ + wire] = s;
#endif
}

// ---------------------------------------------------------------------------
// logits = feats @ fc_w^T + fc_b ; out = log_softmax(logits)
__global__ void k_head(const float* __restrict__ feats,
                       const float* __restrict__ fc_w, const float* __restrict__ fc_b,
                       float* __restrict__ out) {
  int b = threadIdx.x;
  if (b >= BATCH) return;
  float f[NQ];
#pragma unroll
  for (int k = 0; k < NQ; ++k) f[k] = feats[b * NQ + k];
  float lg[NCLS];
  float mx = -3.4e38f;
  for (int c = 0; c < NCLS; ++c) {
    float s = fc_b[c];
#pragma unroll
    for (int k = 0; k < NQ; ++k) s = __builtin_fmaf(f[k], fc_w[c * NQ + k], s);
    lg[c] = s;
    mx = fmaxf(mx, s);
  }
  float sum = 0.0f;
  for (int c = 0; c < NCLS; ++c) sum += expf(lg[c] - mx);
  float lse = mx + logf(sum);
  for (int c = 0; c < NCLS; ++c) out[b * NCLS + c] = lg[c] - lse;
}

// ---------------------------------------------------------------------------
extern "C" void kernel_launch(void* const* d_in, const int* in_sizes, int n_in,
                              void* d_out, int out_size, void* d_ws, size_t ws_size,
                              hipStream_t stream) {
  const float* x    = (const float*)d_in[0];   // (64,16)
  const float* w0   = (const float*)d_in[1];   // (16,3)
  const float* x0   = (const float*)d_in[2];   // (16,)
  const float* w1   = (const float*)d_in[3];   // (16,3)
  const float* x1   = (const float*)d_in[4];   // (16,)
  const float* fc_w = (const float*)d_in[5];   // (23,16)
  const float* fc_b = (const float*)d_in[6];   // (23,)
  float* out = (float*)d_out;                  // (64,23)

  char* ws = (char*)d_ws;
  const size_t STATE_BYTES = (size_t)BATCH * DIM * sizeof(float2);   // 32 MB
  float2* state = (float2*)ws;
  float2* mats  = (float2*)(ws + STATE_BYTES);                       // 64 KB
  float*  crxp  = (float*)(ws + STATE_BYTES + 65536);                // 256 B
  float*  feats = (float*)(ws + STATE_BYTES + 65536 + 256);          // 4 KB

  k_mats<<<8, 256, 0, stream>>>(x, w0, x0, w1, x1, mats, crxp);

  for (int layer = 0; layer < 2; ++layer) {
    k_low1q  <<<2048, 256, 0, stream>>>(state, mats, layer, layer == 0); // (init+) 1q 6..15
    k_high   <<<2048, 256, 0, stream>>>(state, mats, crxp, layer);       // 1q 0..5 + CRX 0..5
    k_lowwrap<<<2048, 256, 0, stream>>>(state, crxp, layer);             // CRX 6..14 + CRX 15
  }

  k_feats<<<BATCH / 16, 256, 0, stream>>>(state, feats);
  k_head<<<1, 64, 0, stream>>>(feats, fc_w, fc_b, out);
}